// BaseSigKernel_8418135900544
// MI455X (gfx1250) — compile-verified
//
#include <hip/hip_runtime.h>

// CDNA5 / gfx1250 signature-kernel:
//   inc = dX * dY^T  via V_WMMA_F32_16X16X4_F32 (K=8 as two chained K=4 WMMAs)
//   Goursat PDE wavefront, one wave32 per (bx,by) pair, lane-shuffle diagonal sweep.

typedef __attribute__((ext_vector_type(2))) float v2f;
typedef __attribute__((ext_vector_type(8))) float v8f;

#define NB     24      // batch per side
#define LX     96      // path length
#define DIM    8       // feature dim (K of the GEMM)
#define TILES  6       // 6*16 = 96 padded inc size (rows/cols 0..94 valid)
#define STRIDE 100     // LDS row stride in floats: lane-halves land 32 banks apart
#define NROWS  190     // PDE interior size = (LX-1)*2^dyadic = 95*2

__global__ __launch_bounds__(32)
void sig_kernel_pde(const float* __restrict__ xs,
                    const float* __restrict__ ys,
                    float* __restrict__ out)
{
    __shared__ float incL[TILES * 16 * STRIDE];  // 96 x 100 floats = 38.4 KB
    __shared__ float edge[NROWS + 2];            // left-boundary column per band

    const int pair = blockIdx.x;            // 0..575
    const int bx   = pair / NB;
    const int by   = pair % NB;
    const int lane = threadIdx.x;           // wave32
    const int half = lane >> 4;             // 0: lanes 0-15, 1: lanes 16-31
    const int l16  = lane & 15;

    // Lane-dependent K offset: lanes 0-15 carry K={0,1}/{4,5}, lanes 16-31 K={2,3}/{6,7}
    const int kb = half * 2;
    const float* __restrict__ X = xs + bx * (LX * DIM) + kb;
    const float* __restrict__ Y = ys + by * (LX * DIM) + kb;
    __builtin_prefetch(X, 0, 3);            // global_prefetch_b8
    __builtin_prefetch(Y, 0, 3);

    // ---------------- Phase 1: a = 0.25 * dX * dY^T via WMMA f32 16x16x4 ----------
    // A fragment (16x4, MxK): lanes 0-15 hold M=l16 with K=(0,1) in (v0,v1);
    // lanes 16-31 K=(2,3).  B (4x16, KxN) mirrors with N=l16.  Each lane loads its
    // two contiguous K-components directly as float2 (no register-select chains).
    v2f a0[TILES], a1[TILES], b0[TILES], b1[TILES];
#pragma unroll
    for (int t = 0; t < TILES; ++t) {
        const int m = t * 16 + l16;                 // increment row index
        v2f A0 = {0.f, 0.f}, A1 = A0, B0 = A0, B1 = A0;
        if (m < LX - 1) {
            const float* xr = X + m * DIM;
            const float* yr = Y + m * DIM;
            const v2f xl0 = *(const v2f*)(xr);
            const v2f xh0 = *(const v2f*)(xr + DIM);
            const v2f xl1 = *(const v2f*)(xr + 4);
            const v2f xh1 = *(const v2f*)(xr + 4 + DIM);
            const v2f yl0 = *(const v2f*)(yr);
            const v2f yh0 = *(const v2f*)(yr + DIM);
            const v2f yl1 = *(const v2f*)(yr + 4);
            const v2f yh1 = *(const v2f*)(yr + 4 + DIM);
            // fold the dyadic 1/(2^q)^2 = 1/4 into A (exact pow2 scale)
            A0 = (xh0 - xl0) * 0.25f;
            A1 = (xh1 - xl1) * 0.25f;
            B0 = yh0 - yl0;
            B1 = yh1 - yl1;
        }
        a0[t] = A0; a1[t] = A1; b0[t] = B0; b1[t] = B1;
    }

#pragma unroll
    for (int ti = 0; ti < TILES; ++ti) {
#pragma unroll
        for (int tj = 0; tj < TILES; ++tj) {
            v8f acc = {};
            acc = __builtin_amdgcn_wmma_f32_16x16x4_f32(
                false, a0[ti], false, b0[tj], (short)0, acc, false, false);
            acc = __builtin_amdgcn_wmma_f32_16x16x4_f32(
                false, a1[ti], false, b1[tj], (short)0, acc, false, false);
            // D layout: lane<16 -> D[r][l16], lane>=16 -> D[r+8][l16]
            const int row0 = ti * 16 + half * 8;
            const int col  = tj * 16 + l16;
#pragma unroll
            for (int r = 0; r < 8; ++r)
                incL[(row0 + r) * STRIDE + col] = acc[r];
        }
    }

    // edge[] = u[i][0] = 1 (left boundary of band 0)
    for (int i = lane; i <= NROWS; i += 32) edge[i] = 1.0f;

    __syncthreads();   // single-wave WG: forces LDS waits before phase 2 reads

    // ---------------- Phase 2: skewed wavefront PDE sweep ----------------
    // u[i][j] = (u[i][j-1] + u[i-1][j]) * c1 - u[i-1][j-1] * c2,
    //   a = incL[(i-1)>>1][(j-1)>>1]   (0.25 already folded in),
    //   c1 = 1 + a/2 + a^2/12,  c2 = 1 - a^2/12,  u[0][*] = u[*][0] = 1.
    const int t = lane;
    for (int blk = 0; blk < TILES; ++blk) {
        const int  j      = blk * 32 + t + 1;       // this lane's column
        const int  jc     = (j - 1) >> 1;           // inc column
        const bool jvalid = (j <= NROWS);
        float up = 1.0f, diag = 1.0f, cur = 1.0f;   // u[i-1][j], u[i-1][j-1], last new
        for (int s = 0; s <= NROWS + 30; ++s) {
            const float lsh = __shfl_up(cur, 1);    // neighbor's u[i][j-1]
            const int   i   = s - t + 1;            // this lane's row this step
            const bool  act = jvalid && (i >= 1) && (i <= NROWS);
            float left = lsh;
            if (t == 0)
                left = edge[(i >= 1 && i <= NROWS) ? i : 0];
            float a = 0.f;
            if (act)
                a = incL[((i - 1) >> 1) * STRIDE + jc];
            const float c1 = 1.0f + a * (0.5f + a * (1.0f / 12.0f));
            const float c2 = 1.0f - a * a * (1.0f / 12.0f);
            const float nv = (left + up) * c1 - diag * c2;
            if (act) {
                diag = left;
                up   = nv;
                cur  = nv;
                if (t == 31) edge[i] = nv;           // right edge -> next band's left
                if (i == NROWS && j == NROWS) out[pair] = nv;  // u[190][190]
            }
        }
    }
}

extern "C" void kernel_launch(void* const* d_in, const int* in_sizes, int n_in,
                              void* d_out, int out_size, void* d_ws, size_t ws_size,
                              hipStream_t stream) {
    const float* xs = (const float*)d_in[0];   // (24, 96, 8) f32
    const float* ys = (const float*)d_in[1];   // (24, 96, 8) f32
    float* out = (float*)d_out;                // (24, 24) f32
    dim3 grid(NB * NB), block(32);
    hipLaunchKernelGGL(sig_kernel_pde, grid, block, 0, stream, xs, ys, out);
}